// RankingLoss_29540785062186
// MI455X (gfx1250) — compile-verified
//
#include <hip/hip_runtime.h>

typedef __attribute__((ext_vector_type(2))) float v2f;
typedef __attribute__((ext_vector_type(8))) float v8f;

#define MARGIN 0.01f
#define NSEL   44      // int(sqrt(2*MAX_PAIRS)) with MAX_PAIRS=1000
#define NPAD   48      // 3 WMMA tiles of 16
#define NSORT  4096
#define BLK    1024

__device__ __forceinline__ unsigned rotl32(unsigned x, int r) {
  return (x << r) | (x >> (32 - r));
}

// Threefry-2x32, 20 rounds (rotations 13,15,26,6 / 17,29,16,24), as used by JAX.
__device__ __forceinline__ void threefry2x32(unsigned k0, unsigned k1,
                                             unsigned c0, unsigned c1,
                                             unsigned &o0, unsigned &o1) {
  const unsigned ks0 = k0, ks1 = k1, ks2 = 0x1BD11BDAu ^ k0 ^ k1;
  unsigned x0 = c0 + ks0, x1 = c1 + ks1;
  const int rotA[4] = {13, 15, 26, 6};
  const int rotB[4] = {17, 29, 16, 24};
#pragma unroll
  for (int g = 0; g < 5; ++g) {
#pragma unroll
    for (int i = 0; i < 4; ++i) {
      int r = (g & 1) ? rotB[i] : rotA[i];
      x0 += x1;
      x1 = rotl32(x1, r);
      x1 ^= x0;
    }
    switch (g) {
      case 0: x0 += ks1; x1 += ks2 + 1u; break;
      case 1: x0 += ks2; x1 += ks0 + 2u; break;
      case 2: x0 += ks0; x1 += ks1 + 3u; break;
      case 3: x0 += ks1; x1 += ks2 + 4u; break;
      default: x0 += ks2; x1 += ks0 + 5u; break;
    }
  }
  o0 = x0;
  o1 = x1;
}

__global__ __launch_bounds__(BLK)
void RankingLoss_29540785062186_kernel(const float* __restrict__ pred,
                                       const float* __restrict__ targ,
                                       float* __restrict__ out) {
  __shared__ unsigned long long s_key[NSORT];  // (bits<<32)|position  -> stable sort
  __shared__ unsigned           s_val[NSORT];  // permuted arange
  __shared__ float              s_ps[NPAD];
  __shared__ float              s_ts[NPAD];

  const int tid = threadIdx.x;

  // ---- Reproduce jax.random.permutation(key(42), 4096): 2 rounds of
  // ---- "sort arange by threefry counter-mode bits" (stable via position LSBs).
  for (int e = tid; e < NSORT; e += BLK) s_val[e] = (unsigned)e;

  unsigned key0 = 0u, key1 = 42u;  // PRNGKey(42) raw data = [0, 42]

  for (int round = 0; round < 2; ++round) {
    // split(key): cipher counters (0,2) and (1,3); row0 = new key, row1 = subkey
    unsigned a02, b02, a13, b13;
    threefry2x32(key0, key1, 0u, 2u, a02, b02);
    threefry2x32(key0, key1, 1u, 3u, a13, b13);
    const unsigned sk0 = b02, sk1 = b13;
    key0 = a02;
    key1 = a13;

    __syncthreads();
    // bits: counters split in halves -> bits[e] = word0(e, e+2048), bits[e+2048] = word1
    for (int e = tid; e < NSORT / 2; e += BLK) {
      unsigned y0, y1;
      threefry2x32(sk0, sk1, (unsigned)e, (unsigned)(e + NSORT / 2), y0, y1);
      s_key[e]             = ((unsigned long long)y0 << 32) | (unsigned)e;
      s_key[e + NSORT / 2] = ((unsigned long long)y1 << 32) | (unsigned)(e + NSORT / 2);
    }
    __syncthreads();

    // bitonic sort of (key,val), ascending; position in low 32 bits => stable
    for (int k = 2; k <= NSORT; k <<= 1) {
      for (int j = k >> 1; j > 0; j >>= 1) {
        for (int i = tid; i < NSORT; i += BLK) {
          const int p = i ^ j;
          if (p > i) {
            const bool asc = ((i & k) == 0);
            const unsigned long long ka = s_key[i], kb = s_key[p];
            if ((ka > kb) == asc) {
              s_key[i] = kb;
              s_key[p] = ka;
              const unsigned va = s_val[i];
              s_val[i] = s_val[p];
              s_val[p] = va;
            }
          }
        }
        __syncthreads();
      }
    }
  }

  // ---- Gather the 44 selected elements, zero-pad to 48 (3 tiles of 16).
  if (tid < NPAD) {
    const bool ok = tid < NSEL;
    const unsigned ix = ok ? s_val[tid] : 0u;
    s_ps[tid] = ok ? pred[ix] : 0.0f;
    s_ts[tid] = ok ? targ[ix] : 0.0f;
  }
  __syncthreads();

  // ---- Wave 0 only (EXEC all ones within the wave, as WMMA requires).
  if ((tid >> 5) == 0) {
    const int lane = tid;        // 0..31
    const int l = lane & 15;
    const bool lo = lane < 16;   // lanes 0-15 carry A cols K=0,1 / B rows 0,1
    const float one = lo ? 1.0f : 0.0f;  // zeroes K=2,3 contribution in hi lanes

    float lsum = 0.0f, lcnt = 0.0f;

#pragma unroll
    for (int I = 0; I < 3; ++I) {
#pragma unroll
      for (int J = 0; J < 3; ++J) {
        if (J < I) continue;  // tiles strictly below diagonal are fully masked
        // diff[m][n] = x_m - x_n via rank-2 WMMA: A=[x | 1], B=[1 ; -x].
        // LDS loads are in-bounds for every lane; mask via multiply (branchless).
        const float rp = one * s_ps[I * 16 + l];
        const float rt = one * s_ts[I * 16 + l];
        const float cp = one * -s_ps[J * 16 + l];
        const float ct = one * -s_ts[J * 16 + l];
        v2f aP = {rp, one};
        v2f bP = {one, cp};
        v2f aT = {rt, one};
        v2f bT = {one, ct};
        v8f z = {};
        v8f pd = __builtin_amdgcn_wmma_f32_16x16x4_f32(false, aP, false, bP,
                                                       (short)0, z, false, false);
        v8f td = __builtin_amdgcn_wmma_f32_16x16x4_f32(false, aT, false, bT,
                                                       (short)0, z, false, false);
        // C/D layout: VGPR r -> M = r (lanes 0-15) or r+8 (lanes 16-31); N = lane&15
        const int nj = J * 16 + l;
#pragma unroll
        for (int r = 0; r < 8; ++r) {
          const int mi = I * 16 + r + (lo ? 0 : 8);
          const float tdv = td[r];
          const float pdv = pd[r];
          // c1/c2 mutually exclusive -> fully branchless select arithmetic
          const float f1 = (tdv > MARGIN) ? 1.0f : 0.0f;   // t_i > t_j + margin
          const float f2 = (tdv < -MARGIN) ? 1.0f : 0.0f;  // t_j > t_i + margin
          const float mk = ((mi < nj) && (nj < NSEL)) ? 1.0f : 0.0f;
          const float r1 = fmaxf(MARGIN - pdv, 0.0f);
          const float r2 = fmaxf(pdv + MARGIN, 0.0f);
          lsum += mk * (f1 * r1 + f2 * r2);
          lcnt += mk * (f1 + f2);
        }
      }
    }

    // wave32 reduction
#pragma unroll
    for (int off = 16; off > 0; off >>= 1) {
      lsum += __shfl_xor(lsum, off, 32);
      lcnt += __shfl_xor(lcnt, off, 32);
    }
    if (lane == 0) out[0] = (lcnt > 0.0f) ? (lsum / lcnt) : 0.0f;
  }
}

extern "C" void kernel_launch(void* const* d_in, const int* in_sizes, int n_in,
                              void* d_out, int out_size, void* d_ws, size_t ws_size,
                              hipStream_t stream) {
  (void)in_sizes; (void)n_in; (void)out_size; (void)d_ws; (void)ws_size;
  const float* pred = (const float*)d_in[0];
  const float* targ = (const float*)d_in[1];
  float* out = (float*)d_out;
  RankingLoss_29540785062186_kernel<<<1, BLK, 0, stream>>>(pred, targ, out);
}